// CorpusAttentivePool_80461917323294
// MI455X (gfx1250) — compile-verified
//
#include <hip/hip_runtime.h>

typedef __attribute__((ext_vector_type(2))) float v2f;
typedef __attribute__((ext_vector_type(8))) float v8f;

#define N_   4
#define C_   256
#define T_   16
#define H_   56
#define W_   56
#define HW_  (H_ * W_)
#define THW_ (T_ * H_ * W_)     // 50176
#define K_   16
#define TO_  8
#define HO_  28
#define WO_  28
#define OVOL_ (TO_ * HO_ * WO_) // 6272

// ---------------------------------------------------------------------------
// Kernel 1: per-batch GEMM w[16x256] x vw[256xTHW] via V_WMMA_F32_16X16X4_F32,
// then max over the 16 words and exp(max/sqrt(C)) -> emap[N,THW].
// One wave computes one 16-position tile; 8 waves / block; EXEC all-ones
// throughout the WMMA loop (tile counts divide exactly).
// ---------------------------------------------------------------------------
__global__ __launch_bounds__(256)
void emap_kernel(const float* __restrict__ vw, const float* __restrict__ w,
                 float* __restrict__ emap) {
  __shared__ float s_w[K_ * C_];   // 16 KB: w[n] as [k][c]

  const int n = blockIdx.y;
  const float* wn = w + (size_t)n * K_ * C_;
  for (int i = threadIdx.x; i < K_ * C_; i += 256) s_w[i] = wn[i];
  __syncthreads();

  const int lane = threadIdx.x & 31;
  const int wid  = threadIdx.x >> 5;
  const int tile = blockIdx.x * 8 + wid;   // 392 * 8 = 3136 tiles = THW/16
  const int pos0 = tile * 16;
  const int m    = lane & 15;              // A: word row / B: position col
  const int kh   = (lane >> 4) << 1;       // channel sub-offset: 0 or 2

  const float* vwn = vw + (size_t)n * C_ * THW_;   // channel-major [C][THW]

  v8f acc = {};
  #pragma unroll 8
  for (int c0 = 0; c0 < C_; c0 += 4) {
    v2f a, b;
    // A-matrix 16x4 (MxK): lane m holds w[m][c0+kh], w[m][c0+kh+1]
    a.x = s_w[m * C_ + c0 + kh];
    a.y = s_w[m * C_ + c0 + kh + 1];
    // B-matrix 4x16 (KxN): lane holds vw[c0+kh][pos0+m], vw[c0+kh+1][pos0+m]
    const float* p = vwn + (size_t)(c0 + kh) * THW_ + (pos0 + m);
    b.x = p[0];
    b.y = p[THW_];
    acc = __builtin_amdgcn_wmma_f32_16x16x4_f32(
        /*neg_a=*/false, a, /*neg_b=*/false, b,
        /*c_mod=*/(short)0, acc, /*reuse_a=*/false, /*reuse_b=*/false);
  }

  // D layout: VGPR r = word M=r (lanes 0-15) / M=8+r (lanes 16-31), col = lane&15.
  float mx = acc[0];
  #pragma unroll
  for (int r = 1; r < 8; ++r) mx = fmaxf(mx, acc[r]);
  mx = fmaxf(mx, __shfl_xor(mx, 16));      // combine M=0..7 with M=8..15
  if (lane < 16)
    emap[(size_t)n * THW_ + pos0 + lane] = expf(mx * 0.0625f);  // 1/sqrt(256)
}

// ---------------------------------------------------------------------------
// Kernel 2: reciprocal denominator: 1 / box3x3x3_stride2_pad1(emap).
// ---------------------------------------------------------------------------
__global__ __launch_bounds__(256)
void denom_kernel(const float* __restrict__ emap, float* __restrict__ rdenom) {
  int i = blockIdx.x * 256 + threadIdx.x;
  if (i >= N_ * OVOL_) return;
  int wo = i % WO_;  int t1 = i / WO_;
  int ho = t1 % HO_; int t2 = t1 / HO_;
  int to = t2 % TO_; int n  = t2 / TO_;

  const float* em = emap + (size_t)n * THW_;
  float s = 0.f;
  for (int kt = 0; kt < 3; ++kt) {
    int t = 2 * to - 1 + kt; if ((unsigned)t >= (unsigned)T_) continue;
    for (int kk = 0; kk < 3; ++kk) {
      int h = 2 * ho - 1 + kk; if ((unsigned)h >= (unsigned)H_) continue;
      int base = (t * H_ + h) * W_;
      for (int kw = 0; kw < 3; ++kw) {
        int ww = 2 * wo - 1 + kw; if ((unsigned)ww >= (unsigned)W_) continue;
        s += em[base + ww];
      }
    }
  }
  rdenom[i] = 1.0f / s;   // s >= 8 positive exp() taps, never zero
}

// ---------------------------------------------------------------------------
// Kernel 3: numerator box filter fused with divide:
//   out[n,c,to,ho,wo] = sum_window(vw * emap) * rdenom[n,to,ho,wo]
// wo fastest across threads -> coalesced output; emap taps stay hot in cache.
// ---------------------------------------------------------------------------
__global__ __launch_bounds__(256)
void out_kernel(const float* __restrict__ vw, const float* __restrict__ emap,
                const float* __restrict__ rdenom, float* __restrict__ out) {
  size_t i = (size_t)blockIdx.x * 256 + threadIdx.x;  // exactly N*C*OVOL threads
  int wo = (int)(i % WO_); size_t t1 = i / WO_;
  int ho = (int)(t1 % HO_); size_t t2 = t1 / HO_;
  int to = (int)(t2 % TO_); size_t t3 = t2 / TO_;
  int c  = (int)(t3 % C_);  int n = (int)(t3 / C_);

  const float* em  = emap + (size_t)n * THW_;
  const float* vwc = vw + ((size_t)n * C_ + c) * THW_;

  float s = 0.f;
  for (int kt = 0; kt < 3; ++kt) {
    int t = 2 * to - 1 + kt; if ((unsigned)t >= (unsigned)T_) continue;
    for (int kk = 0; kk < 3; ++kk) {
      int h = 2 * ho - 1 + kk; if ((unsigned)h >= (unsigned)H_) continue;
      int base = (t * H_ + h) * W_;
      for (int kw = 0; kw < 3; ++kw) {
        int ww = 2 * wo - 1 + kw; if ((unsigned)ww >= (unsigned)W_) continue;
        s = fmaf(vwc[base + ww], em[base + ww], s);
      }
    }
  }
  out[i] = s * rdenom[((n * TO_ + to) * HO_ + ho) * WO_ + wo];
}

// ---------------------------------------------------------------------------
extern "C" void kernel_launch(void* const* d_in, const int* in_sizes, int n_in,
                              void* d_out, int out_size, void* d_ws, size_t ws_size,
                              hipStream_t stream) {
  (void)in_sizes; (void)n_in; (void)out_size; (void)ws_size;

  const float* vw = (const float*)d_in[0];   // [4,256,16,56,56]
  const float* w  = (const float*)d_in[1];   // [4,16,256]
  float* out = (float*)d_out;                // [4,256,8,28,28]

  // workspace layout: emap [N*THW] floats, then rdenom [N*OVOL] floats (~903 KB)
  float* emap   = (float*)d_ws;
  float* rdenom = emap + (size_t)N_ * THW_;

  // 3136 tiles per n, 8 waves per block -> 392 blocks per n
  emap_kernel<<<dim3(THW_ / (16 * 8), N_), 256, 0, stream>>>(vw, w, emap);

  denom_kernel<<<(N_ * OVOL_ + 255) / 256, 256, 0, stream>>>(emap, rdenom);

  // N*C*OVOL = 6,422,528 = 25088 * 256 exactly
  out_kernel<<<(unsigned)((size_t)N_ * C_ * OVOL_ / 256), 256, 0, stream>>>(
      vw, emap, rdenom, out);
}